// HungarianLoss_46059229283296
// MI455X (gfx1250) — compile-verified
//
#include <hip/hip_runtime.h>
#include <hip/hip_bf16.h>
#include <math.h>

typedef __attribute__((ext_vector_type(2))) float v2f;
typedef __attribute__((ext_vector_type(8))) float v8f;
typedef unsigned int u32x4 __attribute__((ext_vector_type(4)));
typedef int i32x4 __attribute__((ext_vector_type(4)));
typedef int i32x8 __attribute__((ext_vector_type(8)));

#define BATCH 128
#define N 256
#define D 64
#define INFF 3.0e38f

#if __has_include(<hip/amd_detail/amd_gfx1250_TDM.h>)
#define TDM_SIX_ARGS 1   // amdgpu-toolchain / therock headers -> 6-arg builtin
#else
#define TDM_SIX_ARGS 0   // ROCm 7.2 -> 5-arg builtin
#endif

// ---------------------------------------------------------------------------
// TDM: DMA a 2D tile (rows x rowlen f32, row stride in elements) into LDS.
// pad68: insert 4 dwords of LDS padding after every 64 dwords (68-float rows).
// Issue from ONE wave; completion via s_wait_tensorcnt.
// ---------------------------------------------------------------------------
__device__ __forceinline__ void tdm_load_2d(unsigned lds_addr, const void* gptr,
                                            unsigned rows, unsigned rowlen,
                                            unsigned stride, bool pad68) {
    unsigned long long ga = (unsigned long long)(size_t)gptr;
    u32x4 g0;
    g0.x = 1u;                                           // count=1 (valid user D#)
    g0.y = lds_addr;                                     // LDS byte address
    g0.z = (unsigned)ga;                                 // global addr lo
    g0.w = (unsigned)((ga >> 32) & 0x01FFFFFFull) | (2u << 30);  // addr hi | type=2
    unsigned d0 = (2u << 16);                            // data_size = 4 bytes
    if (pad68) d0 |= (1u << 20) | (5u << 22) | (3u << 25);  // pad_en, every 64 dw, +4 dw
    i32x8 g1;
    g1[0] = (int)d0;
    g1[1] = (int)(rowlen << 16);                         // tensor_dim0[15:0]
    g1[2] = (int)((rowlen >> 16) | (rows << 16));        // tensor_dim0 hi | tensor_dim1 lo
    g1[3] = (int)((rows >> 16) | (rowlen << 16));        // tensor_dim1 hi | tile_dim0
    g1[4] = (int)rows;                                   // tile_dim1 (tile_dim2 = 0)
    g1[5] = (int)stride;                                 // tensor_dim0_stride lo32
    g1[6] = 0;
    g1[7] = 0;
    i32x4 z4 = {0, 0, 0, 0};
#if TDM_SIX_ARGS
    i32x8 z8 = {0, 0, 0, 0, 0, 0, 0, 0};
    __builtin_amdgcn_tensor_load_to_lds(g0, g1, z4, z4, z8, 0);
#else
    __builtin_amdgcn_tensor_load_to_lds(g0, g1, z4, z4, 0);
#endif
}

// ---------------------------------------------------------------------------
// 1) Per-row inverse L2 norm: one wave per row of 64 floats.
// ---------------------------------------------------------------------------
__global__ __launch_bounds__(256) void rownorm_kernel(const float* __restrict__ x,
                                                      float* __restrict__ inv,
                                                      int rows) {
    int row = blockIdx.x * 8 + (threadIdx.x >> 5);
    if (row >= rows) return;
    int lane = threadIdx.x & 31;
    const float2* p = (const float2*)(x + (size_t)row * D);
    float2 v2 = p[lane];
    float s = v2.x * v2.x + v2.y * v2.y;
#pragma unroll
    for (int m = 16; m >= 1; m >>= 1) s += __shfl_xor(s, m, 32);
    if (lane == 0) {
        float n = sqrtf(s);
        inv[row] = 1.0f / fmaxf(n, 1e-12f);
    }
}

// ---------------------------------------------------------------------------
// 2) Batched GEMM via V_WMMA_F32_16X16X4_F32. Raw panels staged by TDM with
//    hardware LDS padding; normalization folded into the epilogue:
//    C[m][n] = (A.B^T)[m][n] * inv_a[m] * inv_b[n].
// ---------------------------------------------------------------------------
__global__ __launch_bounds__(256) void gemm_wmma_kernel(const float* __restrict__ inputs,
                                                        const float* __restrict__ targets,
                                                        const float* __restrict__ inv_ip,
                                                        const float* __restrict__ inv_tg,
                                                        float* __restrict__ cosm,
                                                        float* __restrict__ selfm) {
    __shared__ float As[64][68];   // 68-float rows produced by TDM pad feature
    __shared__ float Bs[64][68];

    const int b  = blockIdx.y;
    const int m0 = (blockIdx.x >> 2) * 64;
    const int n0 = (blockIdx.x & 3) * 64;
    const float* Bsrc = (blockIdx.z == 0) ? targets : inputs;
    const float* invB = (blockIdx.z == 0) ? inv_tg : inv_ip;
    float*       Cout = (blockIdx.z == 0) ? cosm : selfm;

    if (threadIdx.x < 32) {        // one wave issues both tile DMAs
        tdm_load_2d((unsigned)(size_t)&As[0][0],
                    inputs + ((size_t)b * N + m0) * D, 64, 64, 64, true);
        tdm_load_2d((unsigned)(size_t)&Bs[0][0],
                    Bsrc + ((size_t)b * N + n0) * D, 64, 64, 64, true);
        __builtin_amdgcn_s_wait_tensorcnt(0);
    }
    __syncthreads();

    const int wave = threadIdx.x >> 5;
    const int lane = threadIdx.x & 31;
    const int half = lane >> 4;    // lanes 16-31: K+2 on A/B frags, M+8 on C/D
    const int lm   = lane & 15;

#pragma unroll
    for (int t = wave * 2; t < wave * 2 + 2; ++t) {
        const int tm = t >> 2, tn = t & 3;
        v8f acc = {0.f, 0.f, 0.f, 0.f, 0.f, 0.f, 0.f, 0.f};
#pragma unroll
        for (int kk = 0; kk < 16; ++kk) {
            int k0 = kk * 4 + half * 2;
            v2f a  = *(const v2f*)&As[tm * 16 + lm][k0];
            v2f bb = *(const v2f*)&Bs[tn * 16 + lm][k0];
            acc = __builtin_amdgcn_wmma_f32_16x16x4_f32(
                false, a, false, bb, (short)0, acc, false, false);
        }
        const float sb = invB[(size_t)b * N + n0 + tn * 16 + lm];
        float* c0 = Cout + ((size_t)b * N + m0 + tm * 16 + half * 8) * N + n0 + tn * 16 + lm;
#pragma unroll
        for (int g = 0; g < 8; ++g) {
            float sa = inv_ip[(size_t)b * N + m0 + tm * 16 + half * 8 + g];  // broadcast
            c0[(size_t)g * N] = acc[g] * sa * sb;
        }
    }
}

// ---------------------------------------------------------------------------
// 3) Jonker-Volgenant, one wave32 per batch. Whole 256KB cost matrix is
//    TDM-DMA'd into LDS (320KB/WGP); each lane keeps its 8 owned columns'
//    duals/minv/used in REGISTERS; only u[], p[], way[] (cross-lane) in LDS.
// ---------------------------------------------------------------------------
__global__ __launch_bounds__(32) void hungarian_kernel(const float* __restrict__ cosm,
                                                       int* __restrict__ col4row) {
    __shared__ float cost[N][N];      // raw cos, negated on read (cost = -cos)
    __shared__ float u[N + 1];
    __shared__ int   p[N + 1];
    __shared__ int   way[N + 1];

    const int b = blockIdx.x;
    const int lane = threadIdx.x;

    tdm_load_2d((unsigned)(size_t)&cost[0][0], cosm + (size_t)b * N * N, N, N, N, false);

    for (int j = lane; j <= N; j += 32) { u[j] = 0.f; p[j] = 0; }
    float vloc[8];
#pragma unroll
    for (int s = 0; s < 8; ++s) vloc[s] = 0.f;

    __builtin_amdgcn_s_wait_tensorcnt(0);
    __syncthreads();

    for (int i = 1; i <= N; ++i) {
        if (lane == 0) p[0] = i;
        float minvloc[8];
        int rowid[8];
#pragma unroll
        for (int s = 0; s < 8; ++s) { minvloc[s] = INFF; rowid[s] = 0; }
        unsigned usedmask = 0;
        __syncthreads();

        int j0 = 0;
        int i0 = i;                       // p[0] = i
        for (;;) {
            if (j0 > 0) {                 // owner lane marks column used, caches its row
                int t = j0 - 1;
                if ((t & 31) == lane) { usedmask |= 1u << (t >> 5); rowid[t >> 5] = i0; }
            }
            const float ui0 = u[i0];
            const float* crow = cost[i0 - 1];

            float best = INFF; int bestj = -1;
#pragma unroll
            for (int s = 0; s < 8; ++s) {
                int j = 1 + lane + 32 * s;
                if (!(usedmask & (1u << s))) {
                    float cur = -crow[j - 1] - ui0 - vloc[s];
                    if (cur < minvloc[s]) { minvloc[s] = cur; way[j] = j0; }
                    if (minvloc[s] < best) { best = minvloc[s]; bestj = j; }
                }
            }
            // wave-wide argmin; ties -> smallest column (matches np.argmin)
#pragma unroll
            for (int m = 16; m >= 1; m >>= 1) {
                float ob = __shfl_xor(best, m, 32);
                int   oj = __shfl_xor(bestj, m, 32);
                if (ob < best || (ob == best && oj != -1 && (bestj == -1 || oj < bestj))) {
                    best = ob; bestj = oj;
                }
            }
            const float delta = best;
            const int j1 = bestj;

            // dual updates: used columns' rows get +delta, their v -delta;
            // free columns' minv -delta. Distinct u targets -> race-free.
#pragma unroll
            for (int s = 0; s < 8; ++s) {
                if (usedmask & (1u << s)) { u[rowid[s]] += delta; vloc[s] -= delta; }
                else                        minvloc[s] -= delta;
            }
            if (lane == 0) u[i] += delta;   // virtual column 0 (row i) always used
            __syncthreads();

            j0 = j1;
            i0 = p[j0];
            if (i0 == 0) break;
        }
        if (lane == 0) {                    // augment along alternating path
            int jj = j0;
            while (jj) { int jn = way[jj]; p[jj] = p[jn]; jj = jn; }
        }
        __syncthreads();
    }
    for (int j = 1 + lane; j <= N; j += 32)
        col4row[(size_t)b * N + (p[j] - 1)] = j - 1;
}

// ---------------------------------------------------------------------------
// 4) Output init + row softmax/NLL (one wave per 511-logit row) + global mean.
// ---------------------------------------------------------------------------
__global__ void zero_out_kernel(float* out, int n) {
    int i = blockIdx.x * blockDim.x + threadIdx.x;
    if (i < n) out[i] = 0.f;
}

__global__ __launch_bounds__(256) void nll_kernel(const float* __restrict__ cosm,
                                                  const float* __restrict__ selfm,
                                                  const int* __restrict__ tgt,
                                                  float* __restrict__ out) {
    const int rowid = blockIdx.x * 8 + (threadIdx.x >> 5);
    const int lane  = threadIdx.x & 31;
    const int r     = rowid & (N - 1);
    const float* crow = cosm + (size_t)rowid * N;
    const float* srow = selfm + (size_t)rowid * N;

    float mx = -INFF;
    for (int s = lane; s < N; s += 32) {
        mx = fmaxf(mx, crow[s]);
        if (s != r) mx = fmaxf(mx, srow[s]);     // diagonal of self excluded
    }
#pragma unroll
    for (int m = 16; m >= 1; m >>= 1) mx = fmaxf(mx, __shfl_xor(mx, m, 32));

    float sum = 0.f;
    for (int s = lane; s < N; s += 32) {
        sum += __expf(crow[s] - mx);
        if (s != r) sum += __expf(srow[s] - mx);
    }
#pragma unroll
    for (int m = 16; m >= 1; m >>= 1) sum += __shfl_xor(sum, m, 32);

    if (lane == 0) {
        int t = tgt[rowid];
        float nll = -(crow[t] - mx - __logf(sum));
        atomicAdd(out, nll * (1.0f / (float)(BATCH * N)));
    }
}

// ---------------------------------------------------------------------------
extern "C" void kernel_launch(void* const* d_in, const int* in_sizes, int n_in,
                              void* d_out, int out_size, void* d_ws, size_t ws_size,
                              hipStream_t stream) {
    (void)in_sizes; (void)n_in; (void)ws_size;
    const float* inputs  = (const float*)d_in[0];
    const float* targets = (const float*)d_in[1];
    float* out = (float*)d_out;

    float* cosm   = (float*)d_ws;                                  // [B,N,N]
    float* selfm  = cosm  + (size_t)BATCH * N * N;                 // [B,N,N]
    float* inv_ip = selfm + (size_t)BATCH * N * N;                 // [B*N]
    float* inv_tg = inv_ip + (size_t)BATCH * N;                    // [B*N]
    int*   tgt    = (int*)(inv_tg + (size_t)BATCH * N);            // [B*N]

    const int rows = BATCH * N;                                    // 32768
    rownorm_kernel<<<rows / 8, 256, 0, stream>>>(inputs, inv_ip, rows);
    rownorm_kernel<<<rows / 8, 256, 0, stream>>>(targets, inv_tg, rows);

    dim3 ggrid(16, BATCH, 2);
    gemm_wmma_kernel<<<ggrid, 256, 0, stream>>>(inputs, targets, inv_ip, inv_tg,
                                                cosm, selfm);

    hungarian_kernel<<<BATCH, 32, 0, stream>>>(cosm, tgt);

    zero_out_kernel<<<1, 64, 0, stream>>>(out, out_size);
    nll_kernel<<<rows / 8, 256, 0, stream>>>(cosm, selfm, tgt, out);
}